// GCN_66486093742607
// MI455X (gfx1250) — compile-verified
//
#include <hip/hip_runtime.h>
#include <math.h>

// Problem constants (match reference)
#define B_    4
#define WIN_  5
#define N_    50000
#define E_    1000000
#define H_    32
#define NTOT_ (B_ * N_)   // 200000
#define BE_   (B_ * E_)   // 4000000

typedef __attribute__((ext_vector_type(2))) float v2f;
typedef __attribute__((ext_vector_type(8))) float v8f;

static __device__ __forceinline__ void decode_edge(const int* __restrict__ ei, int e,
                                                   int& src, int& dst) {
    int b  = e / E_;
    int el = e - b * E_;
    src = ei[(b * 2 + 0) * E_ + el] + b * N_;
    dst = ei[(b * 2 + 1) * E_ + el] + b * N_;
}

__global__ void zero_f32_kernel(float* __restrict__ p, int n) {
    int i = blockIdx.x * blockDim.x + threadIdx.x;
    if (i < n) p[i] = 0.0f;
}

__global__ void deg_kernel(const int* __restrict__ ei, float* __restrict__ deg) {
    int e = blockIdx.x * blockDim.x + threadIdx.x;
    if (e >= BE_) return;
    int src, dst;
    decode_edge(ei, e, src, dst);
    atomicAdd(&deg[dst], 1.0f);
}

__global__ void dinv_kernel(float* __restrict__ d) {
    int i = blockIdx.x * blockDim.x + threadIdx.x;
    if (i >= NTOT_) return;
    float v = d[i];
    d[i] = (v > 0.0f) ? rsqrtf(v) : 0.0f;
}

__global__ void norm_kernel(const int* __restrict__ ei, const float* __restrict__ dinv,
                            float* __restrict__ norm) {
    int e = blockIdx.x * blockDim.x + threadIdx.x;
    if (e >= BE_) return;
    int src, dst;
    decode_edge(ei, e, src, dst);
    norm[e] = dinv[src] * dinv[dst];
}

// Narrow propagation (F=5): one thread per (edge, feature)
template <int F>
__global__ void prop_kernel(const int* __restrict__ ei, const float* __restrict__ norm,
                            const float* __restrict__ hin, float* __restrict__ hout) {
    int tid = blockIdx.x * blockDim.x + threadIdx.x;
    int e = tid / F;
    int f = tid - e * F;
    if (e >= BE_) return;
    int src, dst;
    decode_edge(ei, e, src, dst);
    float v = hin[src * F + f] * norm[e];
    atomicAdd(&hout[dst * F + f], v);
}

// Wide propagation (F=32): one thread per (edge, float4 quad) -> b128 gather + 4 f32 atomics
__global__ void prop4_kernel(const int* __restrict__ ei, const float* __restrict__ norm,
                             const float4* __restrict__ hin4, float* __restrict__ hout) {
    int tid = blockIdx.x * blockDim.x + threadIdx.x;
    int e = tid >> 3;          // 8 quads per edge (32 features)
    int q = tid & 7;
    if (e >= BE_) return;
    int src, dst;
    decode_edge(ei, e, src, dst);
    float nrm = norm[e];
    float4 v = hin4[src * 8 + q];
    float* op = &hout[dst * H_ + q * 4];
    atomicAdd(op + 0, v.x * nrm);
    atomicAdd(op + 1, v.y * nrm);
    atomicAdd(op + 2, v.z * nrm);
    atomicAdd(op + 3, v.w * nrm);
}

// out = tanh(concat(x0,x1,x2) @ W + bias), W is (3F x 32) row-major.
// One wave computes a 16x32 output tile via V_WMMA_F32_16X16X4_F32.
template <int F>
__global__ void tag_gemm_kernel(const float* __restrict__ x0, const float* __restrict__ x1,
                                const float* __restrict__ x2, const float* __restrict__ W,
                                const float* __restrict__ bias, float* __restrict__ out) {
    const int numTiles = NTOT_ / 16;                     // 12500
    int wave = threadIdx.x >> 5;
    int tile = blockIdx.x * (blockDim.x >> 5) + wave;
    if (tile >= numTiles) return;                        // wave-uniform: EXEC stays full

    int lane = threadIdx.x & 31;
    int half = lane >> 4;                                // 0: lanes 0-15, 1: lanes 16-31
    int l16  = lane & 15;
    int rowA = tile * 16 + l16;                          // A-matrix row (M)

    v8f c0 = {};  // columns 0..15
    v8f c1 = {};  // columns 16..31

    const float* xs[3] = { x0, x1, x2 };
    constexpr bool FULL = (F % 4 == 0);                  // no K tail -> unguarded loads

#pragma unroll
    for (int s = 0; s < 3; ++s) {
        const float* __restrict__ xp = xs[s];
#pragma unroll
        for (int k0 = 0; k0 < F; k0 += 4) {
            int kA = k0 + half * 2;                      // this lane's K pair within segment
            v2f a, b0, b1;
            if (FULL) {
                // A fragment: 16x4, row = lane&15, half selects K pair 0-1 vs 2-3
                a.x = xp[rowA * F + kA];
                a.y = xp[rowA * F + kA + 1];
                int kW = s * F + kA;                     // row index into W (3F x 32)
                b0.x = W[(kW    ) * H_ + l16];
                b0.y = W[(kW + 1) * H_ + l16];
                b1.x = W[(kW    ) * H_ + 16 + l16];
                b1.y = W[(kW + 1) * H_ + 16 + l16];
            } else {
                // Branch-free K tail: clamp indices in-bounds, zero via arithmetic mask.
                float m0 = (kA     < F) ? 1.0f : 0.0f;
                float m1 = (kA + 1 < F) ? 1.0f : 0.0f;
                int ka0 = (kA     < F) ? kA     : (F - 1);
                int ka1 = (kA + 1 < F) ? kA + 1 : (F - 1);
                a.x = xp[rowA * F + ka0] * m0;
                a.y = xp[rowA * F + ka1] * m1;
                int kw0 = s * F + ka0;
                int kw1 = s * F + ka1;
                b0.x = W[kw0 * H_ + l16] * m0;
                b0.y = W[kw1 * H_ + l16] * m1;
                b1.x = W[kw0 * H_ + 16 + l16] * m0;
                b1.y = W[kw1 * H_ + 16 + l16] * m1;
            }
            c0 = __builtin_amdgcn_wmma_f32_16x16x4_f32(false, a, false, b0,
                                                       (short)0, c0, false, false);
            c1 = __builtin_amdgcn_wmma_f32_16x16x4_f32(false, a, false, b1,
                                                       (short)0, c1, false, false);
        }
    }

    // C/D layout: VGPR r, lanes 0-15 -> M=r, lanes 16-31 -> M=8+r; N = lane&15
#pragma unroll
    for (int r = 0; r < 8; ++r) {
        int row = tile * 16 + half * 8 + r;
        float v0 = tanhf(c0[r] + bias[l16]);
        float v1 = tanhf(c1[r] + bias[16 + l16]);
        out[row * H_ + l16]      = v0;
        out[row * H_ + 16 + l16] = v1;
    }
}

__global__ void readout_kernel(const float* __restrict__ h, const float* __restrict__ Wr,
                               const float* __restrict__ br, float* __restrict__ out) {
    int i = blockIdx.x * blockDim.x + threadIdx.x;
    if (i >= NTOT_) return;
    float acc = br[0];
#pragma unroll
    for (int j = 0; j < H_; ++j) acc += h[i * H_ + j] * Wr[j];
    out[i] = acc;
}

static inline int cdiv(int a, int b) { return (a + b - 1) / b; }

extern "C" void kernel_launch(void* const* d_in, const int* in_sizes, int n_in,
                              void* d_out, int out_size, void* d_ws, size_t ws_size,
                              hipStream_t stream) {
    (void)in_sizes; (void)n_in; (void)out_size; (void)ws_size;

    // Inputs in setup_inputs() dict order
    const float* X  = (const float*)d_in[0];   // (B,WIN,N) flat == (NTOT,5) row-major after reshape
    const int*   EI = (const int*)d_in[1];     // (B,2,E)
    const float* W0 = (const float*)d_in[2];   // (15,32)
    const float* b0 = (const float*)d_in[3];
    const float* W1 = (const float*)d_in[4];   // (96,32)
    const float* b1 = (const float*)d_in[5];
    const float* W2 = (const float*)d_in[6];   // (96,32)
    const float* b2 = (const float*)d_in[7];
    const float* Wr = (const float*)d_in[8];   // (32,1)
    const float* br = (const float*)d_in[9];

    float* outF = (float*)d_out;               // [0,200000) = out ; [200000, +6.4M) = h
    float* hOut = outF + NTOT_;

    // Workspace layout (floats)
    float* ws   = (float*)d_ws;
    float* dinv = ws;                          // NTOT
    float* norm = dinv + NTOT_;                // BE
    float* p1   = norm + BE_;                  // NTOT*32 (layer0 uses first NTOT*5)
    float* p2   = p1 + (size_t)NTOT_ * H_;     // NTOT*32
    float* bufA = p2 + (size_t)NTOT_ * H_;     // NTOT*32
    float* bufB = bufA + (size_t)NTOT_ * H_;   // NTOT*32

    const int TB = 256;

    // ---- gcn_norm ----
    zero_f32_kernel<<<cdiv(NTOT_, TB), TB, 0, stream>>>(dinv, NTOT_);
    deg_kernel<<<cdiv(BE_, TB), TB, 0, stream>>>(EI, dinv);
    dinv_kernel<<<cdiv(NTOT_, TB), TB, 0, stream>>>(dinv);
    norm_kernel<<<cdiv(BE_, TB), TB, 0, stream>>>(EI, dinv, norm);

    const int gemmBlocks = cdiv(NTOT_ / 16, TB / 32);   // 12500 tiles, 8 waves/block
    const int propQ = BE_ * 8;                          // F=32 propagation threads

    // ---- Layer 0: F=5, input is X reinterpreted (NTOT,5) ----
    zero_f32_kernel<<<cdiv(NTOT_ * WIN_, TB), TB, 0, stream>>>(p1, NTOT_ * WIN_);
    prop_kernel<WIN_><<<cdiv(BE_ * WIN_, TB), TB, 0, stream>>>(EI, norm, X, p1);
    zero_f32_kernel<<<cdiv(NTOT_ * WIN_, TB), TB, 0, stream>>>(p2, NTOT_ * WIN_);
    prop_kernel<WIN_><<<cdiv(BE_ * WIN_, TB), TB, 0, stream>>>(EI, norm, p1, p2);
    tag_gemm_kernel<WIN_><<<gemmBlocks, TB, 0, stream>>>(X, p1, p2, W0, b0, bufA);

    // ---- Layer 1: F=32 ----
    zero_f32_kernel<<<cdiv(NTOT_ * H_, TB), TB, 0, stream>>>(p1, NTOT_ * H_);
    prop4_kernel<<<cdiv(propQ, TB), TB, 0, stream>>>(EI, norm, (const float4*)bufA, p1);
    zero_f32_kernel<<<cdiv(NTOT_ * H_, TB), TB, 0, stream>>>(p2, NTOT_ * H_);
    prop4_kernel<<<cdiv(propQ, TB), TB, 0, stream>>>(EI, norm, (const float4*)p1, p2);
    tag_gemm_kernel<H_><<<gemmBlocks, TB, 0, stream>>>(bufA, p1, p2, W1, b1, bufB);

    // ---- Layer 2: F=32, output straight into d_out's h region ----
    zero_f32_kernel<<<cdiv(NTOT_ * H_, TB), TB, 0, stream>>>(p1, NTOT_ * H_);
    prop4_kernel<<<cdiv(propQ, TB), TB, 0, stream>>>(EI, norm, (const float4*)bufB, p1);
    zero_f32_kernel<<<cdiv(NTOT_ * H_, TB), TB, 0, stream>>>(p2, NTOT_ * H_);
    prop4_kernel<<<cdiv(propQ, TB), TB, 0, stream>>>(EI, norm, (const float4*)p1, p2);
    tag_gemm_kernel<H_><<<gemmBlocks, TB, 0, stream>>>(bufB, p1, p2, W2, b2, hOut);

    // ---- Readout ----
    readout_kernel<<<cdiv(NTOT_, TB), TB, 0, stream>>>(hOut, Wr, br, outF);
}